// ChannelEstimator_64905545777647
// MI455X (gfx1250) — compile-verified
//
#include <hip/hip_runtime.h>
#include <hip/hip_bf16.h>

typedef __attribute__((ext_vector_type(2))) float v2f;
typedef __attribute__((ext_vector_type(8))) float v8f;

// ---------------------------------------------------------------------------
// Pass 1: per-pilot weighted LS estimate  H[k] = w[k] * Y[pos[k]] / Xp[k]
// Thread P-1 additionally writes the extrapolated endpoint H[P] (at Nfft-1).
// ---------------------------------------------------------------------------
__global__ void ce_pass1_ls(const float* __restrict__ Yr, const float* __restrict__ Yi,
                            const float* __restrict__ Xpr, const float* __restrict__ Xpi,
                            const float* __restrict__ w, const int* __restrict__ pos,
                            float* __restrict__ Hr, float* __restrict__ Hi,
                            int P, int Nfft) {
    int k = blockIdx.x * blockDim.x + threadIdx.x;
    if (k >= P) return;
    int p = pos[k];
    float yr = Yr[p], yi = Yi[p];
    float xr = Xpr[k], xi = Xpi[k];
    float inv = 1.0f / (xr * xr + xi * xi);
    float wk = w[k];
    float hr = (yr * xr + yi * xi) * inv * wk;
    float hi = (yi * xr - yr * xi) * inv * wk;
    Hr[k] = hr;
    Hi[k] = hi;
    if (k == P - 1) {
        // recompute H[P-2] locally, extrapolate to subcarrier Nfft-1
        int p2 = pos[k - 1];
        float yr2 = Yr[p2], yi2 = Yi[p2];
        float xr2 = Xpr[k - 1], xi2 = Xpi[k - 1];
        float inv2 = 1.0f / (xr2 * xr2 + xi2 * xi2);
        float w2 = w[k - 1];
        float hr2 = (yr2 * xr2 + yi2 * xi2) * inv2 * w2;
        float hi2 = (yi2 * xr2 - yr2 * xi2) * inv2 * w2;
        float dx    = (float)(pos[k] - pos[k - 1]);
        float d_end = (float)(Nfft - 1 - pos[k]);
        Hr[P] = hr + (hr - hr2) / dx * d_end;
        Hi[P] = hi + (hi - hi2) / dx * d_end;
    }
}

// ---------------------------------------------------------------------------
// Pass 2: interpolation as WMMA.  One wave = 16 intervals x 16 offsets = 256
// outputs = one 16x16 f32 D tile:
//   D[m][j] = beta*H[base+m] + alpha*H[base+m+1] + gamma*(j/16)
// A (16x4): rows [H_L, H_R, 1, 0]   B (4x16): [[beta],[alpha],[gamma*j/16],[0]]
// A layout: lanes 0-15 -> row M=lane, K=0,1 ; lanes 16-31 -> row M=lane-16, K=2,3
// B layout: lanes 0-15 -> col N=lane, K=0,1 ; lanes 16-31 -> col N=lane-16, K=2,3
// D layout: VGPR v, lanes 0-15 -> D[v][lane] ; lanes 16-31 -> D[v+8][lane-16]
// ---------------------------------------------------------------------------
__global__ void ce_pass2_wmma(const float* __restrict__ Hr, const float* __restrict__ Hi,
                              const float* __restrict__ pAlpha, const float* __restrict__ pBeta,
                              const float* __restrict__ pGamma,
                              float* __restrict__ out, int P) {
    const float alpha = pAlpha[0];
    const float beta  = pBeta[0];
    const float gamma = pGamma[0];

    int lane  = threadIdx.x & 31;
    int wave  = (blockIdx.x * blockDim.x + threadIdx.x) >> 5;
    int base  = wave * 16;              // first pilot interval of this tile
    bool lo   = lane < 16;
    int  m    = lane & 15;

    v2f a_r, a_i, b_r, b_i;
    if (lo) {
        int t = base + m;               // interval index (t <= P-1, Hr/Hi have P+1)
        a_r.x = Hr[t];  a_r.y = Hr[t + 1];
        a_i.x = Hi[t];  a_i.y = Hi[t + 1];
        b_r.x = beta;   b_r.y = alpha;
        b_i.x = beta;   b_i.y = alpha;
    } else {
        a_r.x = 1.0f;   a_r.y = 0.0f;   // constant columns K=2,3 of A
        a_i.x = 1.0f;   a_i.y = 0.0f;
        b_r.x = gamma * (float)m * 0.0625f;  b_r.y = 0.0f;   // gamma*df row (real only)
        b_i.x = 0.0f;                        b_i.y = 0.0f;
    }

    v8f c = {};
    v8f d_r = __builtin_amdgcn_wmma_f32_16x16x4_f32(false, a_r, false, b_r,
                                                    (short)0, c, false, false);
    v8f d_i = __builtin_amdgcn_wmma_f32_16x16x4_f32(false, a_i, false, b_i,
                                                    (short)0, c, false, false);

    // Store interleaved (real, imag) as float2: each store instruction writes a
    // contiguous 128B segment per half-wave -> fully coalesced b64 stream.
    int rowOff = lo ? 0 : 8;
    int col    = m;
#pragma unroll
    for (int v = 0; v < 8; ++v) {
        int interval = base + rowOff + v;
        int i = interval * 16 + col;    // subcarrier index
        float2 val;
        val.x = d_r[v];
        val.y = d_i[v];
        *reinterpret_cast<float2*>(out + 2 * i) = val;
    }
}

// ---------------------------------------------------------------------------
// Tail fixup: last interval spans 15 subcarriers (df = j/15), overwrite the
// final 16 outputs with the exact values.
// ---------------------------------------------------------------------------
__global__ void ce_fixup_tail(const float* __restrict__ Hr, const float* __restrict__ Hi,
                              const float* __restrict__ pAlpha, const float* __restrict__ pBeta,
                              const float* __restrict__ pGamma,
                              const int* __restrict__ pos,
                              float* __restrict__ out, int P, int Nfft) {
    int j = threadIdx.x;
    if (j >= 16) return;
    float alpha = pAlpha[0], beta = pBeta[0], gamma = pGamma[0];
    float hlr = Hr[P - 1], hli = Hi[P - 1];
    float hrr = Hr[P],     hri = Hi[P];
    float X0 = (float)pos[P - 1];
    float X1 = (float)(Nfft - 1);
    int i = Nfft - 16 + j;
    float df = ((float)i - X0) / (X1 - X0);
    out[2 * i]     = alpha * hrr + beta * hlr + gamma * df;
    out[2 * i + 1] = alpha * hri + beta * hli;
}

extern "C" void kernel_launch(void* const* d_in, const int* in_sizes, int n_in,
                              void* d_out, int out_size, void* d_ws, size_t ws_size,
                              hipStream_t stream) {
    const float* Yr  = (const float*)d_in[0];
    const float* Yi  = (const float*)d_in[1];
    const float* Xpr = (const float*)d_in[2];
    const float* Xpi = (const float*)d_in[3];
    const float* w   = (const float*)d_in[4];
    const float* pa  = (const float*)d_in[5];   // alpha
    const float* pb  = (const float*)d_in[6];   // beta
    const float* pg  = (const float*)d_in[7];   // gamma
    const int*   pos = (const int*)d_in[8];

    int Nfft = in_sizes[0];
    int P    = in_sizes[2];

    float* out = (float*)d_out;
    float* Hr  = (float*)d_ws;                  // P+1 floats (+pad)
    float* Hi  = Hr + (P + 16);

    // Pass 1: pilot LS estimates + extrapolated endpoint.
    ce_pass1_ls<<<(P + 255) / 256, 256, 0, stream>>>(Yr, Yi, Xpr, Xpi, w, pos,
                                                     Hr, Hi, P, Nfft);

    // Pass 2: WMMA interpolation. One wave per 16 intervals; P divisible by 16.
    int waves   = P / 16;
    int threads = waves * 32;
    ce_pass2_wmma<<<(threads + 255) / 256, 256, 0, stream>>>(Hr, Hi, pa, pb, pg,
                                                             out, P);

    // Tail: exact df = j/15 for the final (short) interval.
    ce_fixup_tail<<<1, 16, 0, stream>>>(Hr, Hi, pa, pb, pg, pos, out, P, Nfft);
}